// SpectralMoleculeEncoder_70351564308694
// MI455X (gfx1250) — compile-verified
//
#include <hip/hip_runtime.h>
#include <hip/hip_bf16.h>

#define N_NODES 10000
#define N_EDGES 160000

typedef __attribute__((ext_vector_type(2))) float v2f;
typedef __attribute__((ext_vector_type(8))) float v8f;

// ---------------- graph-normalization kernels ----------------

__global__ void deg_kernel(const int* __restrict__ row, float* __restrict__ deg, int ne) {
    int e = blockIdx.x * blockDim.x + threadIdx.x;
    if (e < ne) atomicAdd(&deg[row[e]], 1.0f);
}

__global__ void dis_kernel(const float* __restrict__ deg, float* __restrict__ dis, int n) {
    int i = blockIdx.x * blockDim.x + threadIdx.x;
    if (i < n) {
        float d = deg[i];
        dis[i] = (d > 0.0f) ? (1.0f / sqrtf(d)) : 0.0f;
    }
}

__global__ void ew_kernel(const int* __restrict__ row, const int* __restrict__ col,
                          const float* __restrict__ dis, float* __restrict__ ew, int ne) {
    int e = blockIdx.x * blockDim.x + threadIdx.x;
    if (e < ne) ew[e] = -dis[row[e]] * dis[col[e]];
}

// out[col[e]*F + f] += ew[e] * z[row[e]*F + f]   (blockDim.x == F)
__global__ void prop_scatter(const float* __restrict__ z, const int* __restrict__ row,
                             const int* __restrict__ col, const float* __restrict__ ew,
                             float* __restrict__ out) {
    const int e = blockIdx.x;
    const int F = blockDim.x;
    const int r = row[e];     // uniform -> scalar load
    const int c = col[e];
    const float wv = ew[e];
    const int f = threadIdx.x;
    atomicAdd(&out[(size_t)c * F + f], wv * z[(size_t)r * F + f]);
}

// t2 = 2*t2 - t0
__global__ void axpby_kernel(float* __restrict__ t2, const float* __restrict__ t0, int n) {
    int i = blockIdx.x * blockDim.x + threadIdx.x;
    if (i < n) t2[i] = 2.0f * t2[i] - t0[i];
}

// ---------------- fp32 WMMA Chebyshev GEMM ----------------
// One wave computes a 16(M) x 64(N) tile of out = sum_t T_t @ W[t] + bias (+ReLU).
// A fragment (16x4 f32): lanes 0-15 hold K=kk..kk+1 with kk=0 ; lanes 16-31 kk=2 (M = lane&15).
// B fragment (4x16 f32): mirrored layout with N = lane&15.
// C/D (16x16 f32, 8 VGPRs): VGPR v -> rows v (lanes 0-15) and v+8 (lanes 16-31).
// One A fragment is reused by 4 WMMAs (4 adjacent N-tiles) -> 4x arithmetic intensity on A.
__global__ __launch_bounds__(32)
void cheb_gemm(const float* __restrict__ T0, const float* __restrict__ T1,
               const float* __restrict__ T2, const float* __restrict__ W,
               const float* __restrict__ bias, float* __restrict__ out,
               int Fin, int Fout, int nterms, int relu) {
    const int lane  = threadIdx.x;
    const int m16   = lane & 15;
    const int half  = lane >> 4;           // 0 or 1
    const int kk    = half << 1;           // 0 or 2
    const int mrow  = (blockIdx.x << 4) + m16;
    const int ncol0 = (blockIdx.y << 6) + m16;   // base of 64-wide N tile

    v8f acc0 = {}, acc1 = {}, acc2 = {}, acc3 = {};
    for (int t = 0; t < nterms; ++t) {
        const float* arow = ((t == 0) ? T0 : (t == 1) ? T1 : T2)
                            + (size_t)mrow * Fin + kk;
        const float* B = W + (size_t)t * Fin * Fout + ncol0;
        #pragma unroll 2
        for (int k = 0; k < Fin; k += 4) {
            v2f a = *(const v2f*)(arow + k);                 // A[m][k+kk], A[m][k+kk+1]
            const float* Brow0 = B + (size_t)(k + kk) * Fout; // W row k+kk
            const float* Brow1 = Brow0 + Fout;                // W row k+kk+1
            v2f b0, b1, b2, b3;
            b0.x = Brow0[0];   b0.y = Brow1[0];
            b1.x = Brow0[16];  b1.y = Brow1[16];
            b2.x = Brow0[32];  b2.y = Brow1[32];
            b3.x = Brow0[48];  b3.y = Brow1[48];
            acc0 = __builtin_amdgcn_wmma_f32_16x16x4_f32(false, a, false, b0, (short)0, acc0, false, false);
            acc1 = __builtin_amdgcn_wmma_f32_16x16x4_f32(false, a, false, b1, (short)0, acc1, false, false);
            acc2 = __builtin_amdgcn_wmma_f32_16x16x4_f32(false, a, false, b2, (short)0, acc2, false, false);
            acc3 = __builtin_amdgcn_wmma_f32_16x16x4_f32(false, a, false, b3, (short)0, acc3, false, false);
        }
    }

    const int rbase = (blockIdx.x << 4) + (half << 3);  // +0 or +8
    const float bv0 = bias[ncol0];
    const float bv1 = bias[ncol0 + 16];
    const float bv2 = bias[ncol0 + 32];
    const float bv3 = bias[ncol0 + 48];
    #pragma unroll
    for (int v = 0; v < 8; ++v) {
        float* orow = out + (size_t)(rbase + v) * Fout + (size_t)ncol0;
        float x0 = acc0[v] + bv0;
        float x1 = acc1[v] + bv1;
        float x2 = acc2[v] + bv2;
        float x3 = acc3[v] + bv3;
        if (relu) {
            x0 = fmaxf(x0, 0.0f); x1 = fmaxf(x1, 0.0f);
            x2 = fmaxf(x2, 0.0f); x3 = fmaxf(x3, 0.0f);
        }
        orow[0]  = x0;
        orow[16] = x1;
        orow[32] = x2;
        orow[48] = x3;
    }
}

// ---------------- host-side orchestration ----------------

extern "C" void kernel_launch(void* const* d_in, const int* in_sizes, int n_in,
                              void* d_out, int out_size, void* d_ws, size_t ws_size,
                              hipStream_t stream) {
    const float* v   = (const float*)d_in[0];
    const int*   ei  = (const int*)  d_in[1];
    const float* W1  = (const float*)d_in[2];
    const float* b1  = (const float*)d_in[3];
    const float* W2  = (const float*)d_in[4];
    const float* b2  = (const float*)d_in[5];
    const float* W3  = (const float*)d_in[6];
    const float* b3  = (const float*)d_in[7];
    const float* Wmu = (const float*)d_in[8];
    const float* bmu = (const float*)d_in[9];
    const float* Wsd = (const float*)d_in[10];
    const float* bsd = (const float*)d_in[11];

    const int* row = ei;             // edge_index[0]
    const int* col = ei + N_EDGES;   // edge_index[1]

    // workspace layout (floats)
    float* ws  = (float*)d_ws;
    float* deg = ws;                          ws += 10240;
    float* dis = ws;                          ws += 10240;
    float* ew  = ws;                          ws += N_EDGES;
    float* tx1 = ws;                          ws += (size_t)N_NODES * 256;
    float* tx2 = ws;                          ws += (size_t)N_NODES * 256;
    float* x1  = ws;                          ws += (size_t)N_NODES * 128;
    float* x2  = ws;                          ws += (size_t)N_NODES * 256;
    float* x3  = ws;                          ws += (size_t)N_NODES * 512;

    float* mu = (float*)d_out;
    float* sd = mu + (size_t)N_NODES * 256;

    // normalized edge weights: w = -deg[row]^{-1/2} * deg[col]^{-1/2}
    hipMemsetAsync(deg, 0, N_NODES * sizeof(float), stream);
    deg_kernel<<<(N_EDGES + 255) / 256, 256, 0, stream>>>(row, deg, N_EDGES);
    dis_kernel<<<(N_NODES + 255) / 256, 256, 0, stream>>>(deg, dis, N_NODES);
    ew_kernel <<<(N_EDGES + 255) / 256, 256, 0, stream>>>(row, col, dis, ew, N_EDGES);

    auto run_layer = [&](const float* xin, int Fin, int Fout,
                         const float* W, const float* b, float* xout) {
        const size_t nin = (size_t)N_NODES * (size_t)Fin;
        // tx1 = L_hat @ xin
        hipMemsetAsync(tx1, 0, nin * sizeof(float), stream);
        prop_scatter<<<N_EDGES, Fin, 0, stream>>>(xin, row, col, ew, tx1);
        // tx2 = 2 * L_hat @ tx1 - xin
        hipMemsetAsync(tx2, 0, nin * sizeof(float), stream);
        prop_scatter<<<N_EDGES, Fin, 0, stream>>>(tx1, row, col, ew, tx2);
        axpby_kernel<<<((int)nin + 255) / 256, 256, 0, stream>>>(tx2, xin, (int)nin);
        // xout = relu(xin@W0 + tx1@W1 + tx2@W2 + b)
        dim3 g(N_NODES / 16, Fout / 64);
        cheb_gemm<<<g, 32, 0, stream>>>(xin, tx1, tx2, W, b, xout, Fin, Fout, 3, 1);
    };

    run_layer(v,  128, 128, W1, b1, x1);
    run_layer(x1, 128, 256, W2, b2, x2);
    run_layer(x2, 256, 512, W3, b3, x3);

    // heads: mu = x3 @ Wmu + bmu ; std = x3 @ Wstd + bstd  (no relu)
    dim3 gh(N_NODES / 16, 256 / 64);
    cheb_gemm<<<gh, 32, 0, stream>>>(x3, x3, x3, Wmu, bmu, mu, 512, 256, 1, 0);
    cheb_gemm<<<gh, 32, 0, stream>>>(x3, x3, x3, Wsd, bsd, sd, 512, 256, 1, 0);
}